// CapsuleLayer_69681549410341
// MI455X (gfx1250) — compile-verified
//
#include <hip/hip_runtime.h>
#include <math.h>

// Problem constants (from reference setup_inputs)
#define B_    32
#define IN_   2048
#define DIN_  8
#define N_    64
#define DOUT_ 16
#define C_    32
#define REP_  (IN_ / C_)     // 64 consecutive i share one channel's weights
#define TILE_ 512            // i-tile held in LDS (32 KB of hat)
#define NTILE (IN_ / TILE_)
#define EPS_  1e-7f

typedef float v2f __attribute__((ext_vector_type(2)));
typedef float v8f __attribute__((ext_vector_type(8)));

// Compute one i-tile of vote vectors hat[i_local, d] = sum_k x[b,i,k] * W[n, i/64, d, k]
// into LDS using V_WMMA_F32_16X16X4_F32 (M=16 rows of i, N=16 cols of d, K=8 via 2 chained WMMAs).
//
// A (16x4 f32) layout: lanes 0-15 hold K={0,1} in vgpr{0,1}; lanes 16-31 hold K={2,3}.
// B (4x16 f32) layout: lanes carry N=d; lane-half selects K pair (mirror of A).
// D (16x16 f32) layout: vgpr r, lanes 0-15 -> M=r; lanes 16-31 -> M=r+8; N = lane&15.
__device__ __forceinline__ void hat_tile_wmma(
    const float* __restrict__ x, const float* __restrict__ Wt,
    float* __restrict__ hat, int b, int n, int tile, int wave, int lane)
{
  const int mrow = lane & 15;          // A: M row index / B: N (=d) column index
  const int kh   = (lane >> 4) << 1;   // K half select: 0 or 2
  const int half = lane >> 4;
  // 32 tile-units of 16 i-rows each; 8 waves -> 4 units per wave, 8 WMMAs/wave.
  for (int u = wave; u < TILE_ / 16; u += 8) {
    const int il0 = u << 4;                  // local i base within tile
    const int gi0 = tile * TILE_ + il0;      // global i base
    const int cu  = gi0 >> 6;                // channel = i / 64  (jnp.repeat semantics)
    const float* xr = x  + ((size_t)b * IN_ + gi0 + mrow) * DIN_ + kh;
    const float* wr = Wt + (((size_t)n * C_ + cu) * DOUT_ + mrow) * DIN_ + kh;
    v2f a0 = { xr[0], xr[1] };   // K = kh..kh+1   (first  K-group 0..3)
    v2f a1 = { xr[4], xr[5] };   // K = kh+4..kh+5 (second K-group 4..7)
    v2f b0 = { wr[0], wr[1] };   // B[k][d] = W[n][cu][d][k]
    v2f b1 = { wr[4], wr[5] };
    v8f acc = {0.f,0.f,0.f,0.f,0.f,0.f,0.f,0.f};
    acc = __builtin_amdgcn_wmma_f32_16x16x4_f32(false, a0, false, b0,
                                                (short)0, acc, false, false);
    acc = __builtin_amdgcn_wmma_f32_16x16x4_f32(false, a1, false, b1,
                                                (short)0, acc, false, false);
#pragma unroll
    for (int r = 0; r < 8; ++r)
      hat[(il0 + r + (half << 3)) * DOUT_ + mrow] = acc[r];
  }
}

// Per-(b,i) softmax statistics over the n axis: max and 1/sum(exp).
__global__ __launch_bounds__(256) void caps_softmax(
    const float* __restrict__ logits, float* __restrict__ smax, float* __restrict__ sinv)
{
  const int e = blockIdx.x * 256 + threadIdx.x;   // e in [0, B_*IN_)
  const int b = e >> 11;                           // / IN_
  const int i = e & (IN_ - 1);
  const float* lp = logits + (size_t)b * N_ * IN_ + i;
  float mx = -3.4e38f;
#pragma unroll 4
  for (int n = 0; n < N_; ++n) mx = fmaxf(mx, lp[(size_t)n * IN_]);
  float s = 0.f;
#pragma unroll 4
  for (int n = 0; n < N_; ++n) s += __expf(lp[(size_t)n * IN_] - mx);
  smax[e] = mx;
  sinv[e] = 1.0f / s;
}

// One routing iteration for one (b,n) pair per workgroup.
// MODE 0: c = 1/N uniform, logits = delta (first write)
// MODE 1: c = softmax(logits), logits += delta
// MODE 2: c = softmax(logits), final -> only writes outputs
template <int MODE>
__global__ __launch_bounds__(256) void caps_route(
    const float* __restrict__ x, const float* __restrict__ Wt,
    const float* __restrict__ bias, float* __restrict__ logits,
    const float* __restrict__ smax, const float* __restrict__ sinv,
    float* __restrict__ out)
{
  __shared__ float hat[TILE_ * DOUT_];   // 32 KB vote tile
  __shared__ float cld[TILE_];           // routing coefficients for tile
  __shared__ float part[256];            // s partial sums
  __shared__ float sld[DOUT_];
  __shared__ float old_[DOUT_];
  __shared__ float scl;

  const int b    = blockIdx.x >> 6;        // / N_
  const int n    = blockIdx.x & (N_ - 1);
  const int wave = threadIdx.x >> 5;
  const int lane = threadIdx.x & 31;
  const int dc    = threadIdx.x & 15;
  const int chunk = threadIdx.x >> 4;      // 16 chunks x 32 i's per tile
  const size_t lbase = ((size_t)b * N_ + n) * IN_;

  // ---- Pass 1: s[d] = sum_i c[i] * hat[i,d] ----
  float sacc = 0.f;
  for (int tile = 0; tile < NTILE; ++tile) {
    if (tile + 1 < NTILE)   // hint next x tile toward the caches (global_prefetch_b8)
      __builtin_prefetch(
          x + ((size_t)b * IN_ + (tile + 1) * TILE_ + (threadIdx.x << 1)) * DIN_, 0, 0);
    hat_tile_wmma(x, Wt, hat, b, n, tile, wave, lane);
    for (int ii = threadIdx.x; ii < TILE_; ii += 256) {
      float cv;
      if (MODE == 0) {
        cv = 1.0f / (float)N_;
      } else {
        const int gi = tile * TILE_ + ii;
        const size_t e = (size_t)b * IN_ + gi;
        cv = __expf(logits[lbase + gi] - smax[e]) * sinv[e];
      }
      cld[ii] = cv;
    }
    __syncthreads();
    const float* hp = hat + (chunk * 32) * DOUT_ + dc;
    const float* cp = cld + chunk * 32;
#pragma unroll 8
    for (int ii = 0; ii < 32; ++ii) sacc += cp[ii] * hp[ii * DOUT_];
    __syncthreads();
  }

  // ---- Reduce partials, add bias, squash ----
  part[threadIdx.x] = sacc;
  __syncthreads();
  if (threadIdx.x < DOUT_) {
    float s = bias[n * DOUT_ + threadIdx.x];
#pragma unroll
    for (int ch = 0; ch < 16; ++ch) s += part[ch * 16 + threadIdx.x];
    sld[threadIdx.x] = s;
  }
  __syncthreads();
  if (threadIdx.x == 0) {
    float sn = 0.f;
#pragma unroll
    for (int d2 = 0; d2 < DOUT_; ++d2) sn += sld[d2] * sld[d2];
    scl = sn / (1.0f + sn) * rsqrtf(sn + EPS_);   // sn/(1+sn)/sqrt(sn+eps)
  }
  __syncthreads();
  if (threadIdx.x < DOUT_) {
    const float o = scl * sld[threadIdx.x];
    old_[threadIdx.x] = o;
    out[((size_t)b * N_ + n) * DOUT_ + threadIdx.x] = o;
  }
  __syncthreads();

  // ---- Pass 2: logits update delta[i] = sum_d out[d]*hat[i,d] (recompute hat) ----
  if (MODE < 2) {
    for (int tile = 0; tile < NTILE; ++tile) {
      hat_tile_wmma(x, Wt, hat, b, n, tile, wave, lane);
      __syncthreads();
      for (int ii = threadIdx.x; ii < TILE_; ii += 256) {
        const float* hr = hat + ii * DOUT_;
        float dlt = 0.f;
#pragma unroll
        for (int d2 = 0; d2 < DOUT_; ++d2) dlt += old_[d2] * hr[d2];
        const size_t g = lbase + tile * TILE_ + ii;
        logits[g] = (MODE == 0) ? dlt : (logits[g] + dlt);
      }
      __syncthreads();
    }
  }
}

extern "C" void kernel_launch(void* const* d_in, const int* in_sizes, int n_in,
                              void* d_out, int out_size, void* d_ws, size_t ws_size,
                              hipStream_t stream)
{
  const float* x    = (const float*)d_in[0];   // [B, In, Din]
  const float* Wt   = (const float*)d_in[1];   // [N, C, Dout, Din]
  const float* bias = (const float*)d_in[2];   // [N, Dout]
  float* out = (float*)d_out;                  // [B, N, Dout]

  // Workspace layout (all f32): logits[B,N,In] | smax[B,In] | sinv[B,In]  (~16.5 MB, L2-resident)
  float* logits = (float*)d_ws;
  float* smax   = logits + (size_t)B_ * N_ * IN_;
  float* sinv   = smax   + (size_t)B_ * IN_;

  dim3 grid(B_ * N_), blk(256);
  dim3 sgrid((B_ * IN_) / 256), sblk(256);

  // iter 0 (uniform c) -> stats -> iter 1 -> stats -> iter 2 (final outputs)
  caps_route<0><<<grid, blk, 0, stream>>>(x, Wt, bias, logits, smax, sinv, out);
  caps_softmax<<<sgrid, sblk, 0, stream>>>(logits, smax, sinv);
  caps_route<1><<<grid, blk, 0, stream>>>(x, Wt, bias, logits, smax, sinv, out);
  caps_softmax<<<sgrid, sblk, 0, stream>>>(logits, smax, sinv);
  caps_route<2><<<grid, blk, 0, stream>>>(x, Wt, bias, logits, smax, sinv, out);
}